// LengthRegulator_24129126269102
// MI455X (gfx1250) — compile-verified
//
#include <hip/hip_runtime.h>

// ---------------------------------------------------------------------------
// MI455X (gfx1250) implementation.
//   conv1d(K=3,'same') == GEMM [16384 x 768] @ [768 x 256] with free im2col.
//   f32 WMMA (V_WMMA_F32_16X16X4_F32) keeps f32-class precision vs reference.
//   LDS staging uses the CDNA5 async/tensor data movers, double-buffered:
//     - A tile: GLOBAL_LOAD_ASYNC_TO_LDS_B128 (ASYNCcnt), per-lane scatter,
//       boundary rows zero-filled with ds_store on disjoint lanes.
//     - W tile: one TDM tensor_load_to_lds per step (TENSORcnt), with
//       pad_enable reproducing the bank-padded LDS row stride in hardware.
//   Epilogue fuses bias + LayerNorm + ReLU (and the 256->1 linear in phase 2)
//   so h2 never touches HBM. Expand is bandwidth-bound (128 MiB) -> b128
//   nontemporal streaming stores.
// ---------------------------------------------------------------------------

typedef float v2f  __attribute__((ext_vector_type(2)));
typedef float v8f  __attribute__((ext_vector_type(8)));
typedef float f4   __attribute__((ext_vector_type(4)));
typedef unsigned int u32x4 __attribute__((ext_vector_type(4)));
typedef int  i32x4 __attribute__((ext_vector_type(4)));
typedef int  i32x8 __attribute__((ext_vector_type(8)));

// The async-LDS builtins take pointers to gcc-style int 4-vectors (V4i) in
// AS1 (global / __device__) and AS3 (LDS / __shared__).
typedef int gvi4 __attribute__((vector_size(16)));
typedef __attribute__((address_space(1))) gvi4 glb_i4;
typedef __attribute__((address_space(3))) gvi4 lds_i4;

#define BATCH 32
#define SEQL  512
#define CH    256
#define NKSTEP 24            // 768 / 32
#define LDA 36               // 32 + 4 pad: 16B-aligned rows, 16 distinct banks
#define LDB 264              // 256 + 8 pad
#define LDO 264
#define A_FLOATS (64 * LDA)  // 2304
#define B_FLOATS (32 * LDB)  // 8448
#define BUF (A_FLOATS + B_FLOATS)   // 10752 floats per pipeline stage

#if __has_builtin(__builtin_amdgcn_global_load_async_to_lds_b128)
#define HAVE_ASYNC 1
#else
#define HAVE_ASYNC 0
#endif
#if __has_builtin(__builtin_amdgcn_tensor_load_to_lds)
#define HAVE_TDM 1
#else
#define HAVE_TDM 0
#endif

#if __has_builtin(__builtin_amdgcn_s_wait_asynccnt)
#define WAIT_ASYNC(n) __builtin_amdgcn_s_wait_asynccnt(n)
#else
#define WAIT_ASYNC(n) asm volatile("s_wait_asynccnt %0" ::"i"(n) : "memory")
#endif
#if __has_builtin(__builtin_amdgcn_s_wait_tensorcnt)
#define WAIT_TENSOR(n) __builtin_amdgcn_s_wait_tensorcnt(n)
#else
#define WAIT_TENSOR(n) asm volatile("s_wait_tensorcnt %0" ::"i"(n) : "memory")
#endif

static __device__ __forceinline__ glb_i4* to_glb(const void* p) {
  return (glb_i4*)(unsigned long long)p;        // int->ptr: no AS-cast needed
}
static __device__ __forceinline__ lds_i4* to_lds(const void* p) {
  // generic shared pointer: low 32 bits are the LDS byte offset (ISA 10.2)
  return (lds_i4*)(unsigned)(unsigned long long)p;
}

static __device__ __forceinline__ v2f mk2(float a, float b) { v2f r; r.x = a; r.y = b; return r; }

// PHASE 1: in = x,  writes h1 = relu(LN(conv1(x)))
// PHASE 2: in = h1, writes dur_pred = relu(relu(LN(conv2(h1))) @ lw + lb)
template <int PHASE>
__global__ __launch_bounds__(256) void conv_ln_kernel(
    const float* __restrict__ in,    // [16384, 256]
    const float* __restrict__ wmat,  // [768, 256]
    const float* __restrict__ cb,    // [256] conv bias
    const float* __restrict__ g,     // [256] LN gamma
    const float* __restrict__ beta,  // [256] LN beta
    const float* __restrict__ lw,    // [256] (phase 2)
    const float* __restrict__ lbp,   // [1]   (phase 2)
    float* __restrict__ out_h,       // [16384, 256] (phase 1)
    float* __restrict__ out_d)       // [16384]      (phase 2)
{
  __shared__ float smem[2 * BUF];           // 86 KB: two pipeline stages

  const int tid    = threadIdx.x;
  const int lane   = tid & 31;
  const int wv     = tid >> 5;              // 8 waves
  const int wm     = wv >> 2;               // 0..1  (M direction, 32 rows each)
  const int wn     = wv & 3;                // 0..3  (N direction, 64 cols each)
  const int hi     = lane >> 4;             // lane half (K-pair select)
  const int l16    = lane & 15;
  const int m_base = blockIdx.x * 64;

  // outstanding async/tensor ops issued per wave per pipeline stage
  constexpr int NASY = HAVE_ASYNC ? (HAVE_TDM ? 2 : 10) : 0;
  constexpr int NTEN = HAVE_TDM ? 1 : 0;
  (void)NASY; (void)NTEN;

  v8f c[2][4];
#pragma unroll
  for (int mi = 0; mi < 2; ++mi)
#pragma unroll
    for (int ni = 0; ni < 4; ++ni)
      c[mi][ni] = (v8f){0.f,0.f,0.f,0.f,0.f,0.f,0.f,0.f};

  // ---- stage copy: global -> LDS via async/TDM movers --------------------
  auto issue_copy = [&](int ks, int p) {
    const int k_base = ks * 32;
    float* A_l = smem + p * BUF;
    float* B_l = A_l + A_FLOATS;

    // A tile: 64 rows x 32 cols of im2col(x): j -> x[b, l-1 + j/256, j%256]
    {
      const int mr  = tid >> 2;
      const int kc  = (tid & 3) * 8;
      const int m   = m_base + mr;
      const int l   = m & (SEQL - 1);
      const int bI  = m >> 9;
      const int j   = k_base + kc;
      const int r   = l - 1 + (j >> 8);
      const int col = j & 255;
      const bool valid = (unsigned)r < (unsigned)SEQL;
      const float* src = in + (((size_t)(bI << 9) + (size_t)(valid ? r : 0)) << 8) + col;
      float* dst = A_l + mr * LDA + kc;     // 16B aligned (LDA*4 % 16 == 0)
#if HAVE_ASYNC
      if (valid) {
        __builtin_amdgcn_global_load_async_to_lds_b128(to_glb(src),     to_lds(dst),     0, 0);
        __builtin_amdgcn_global_load_async_to_lds_b128(to_glb(src + 4), to_lds(dst + 4), 0, 0);
      } else {
        const f4 z = (f4){0.f,0.f,0.f,0.f};
        ((f4*)dst)[0] = z; ((f4*)dst)[1] = z;
      }
#else
      f4 a0 = (f4){0.f,0.f,0.f,0.f}, a1 = a0;
      if (valid) { a0 = ((const f4*)src)[0]; a1 = ((const f4*)src)[1]; }
      ((f4*)dst)[0] = a0; ((f4*)dst)[1] = a1;
#endif
    }

    // W tile: 32 x 256 contiguous rows of [768,256]
#if HAVE_TDM
    if (wv == 0) {
      const unsigned lds_off = (unsigned)(unsigned long long)(const void*)B_l;
      const unsigned long long ga =
          (unsigned long long)(const void*)(wmat + (size_t)k_base * CH);
      u32x4 g0;
      g0[0] = 1u;                                    // count=1, user descriptor
      g0[1] = lds_off;                               // lds_addr (bytes)
      g0[2] = (unsigned)ga;                          // global_addr[31:0]
      g0[3] = (unsigned)((ga >> 32) & 0x01FFFFFFu) | (2u << 30);  // addr[56:32], type=2
      i32x8 g1;
      g1[0] = (2 << 16)        // data_size = 4B
            | (1 << 20)        // pad_enable
            | (7 << 22)        // pad_interval: every 256 dwords
            | (7 << 25);       // pad_amount: 8 dwords  -> LDS row stride 264
      g1[1] = (int)(256u << 16);   // tensor_dim0 = 256 (atomic_barrier_addr = 0)
      g1[2] = (int)(32u << 16);    // tensor_dim1 = 32
      g1[3] = (int)(256u << 16);   // tile_dim0 = 256
      g1[4] = 32;                  // tile_dim1 = 32, tile_dim2 = 0 (2D)
      g1[5] = 256;                 // tensor_dim0_stride = 256
      g1[6] = 0;
      g1[7] = 0;
      const i32x4 z4 = (i32x4){0, 0, 0, 0};
#if __clang_major__ >= 23
      const i32x8 z8 = (i32x8){0, 0, 0, 0, 0, 0, 0, 0};
      __builtin_amdgcn_tensor_load_to_lds(g0, g1, z4, z4, z8, 0);
#else
      __builtin_amdgcn_tensor_load_to_lds(g0, g1, z4, z4, 0);
#endif
    }
#elif HAVE_ASYNC
    {
      const int kc = tid >> 3;
      const int nb = (tid & 7) * 32;
      const float* src = wmat + (size_t)(k_base + kc) * CH + nb;
      float* dst = B_l + kc * LDB + nb;
#pragma unroll
      for (int i = 0; i < 8; ++i)
        __builtin_amdgcn_global_load_async_to_lds_b128(to_glb(src + 4 * i),
                                                       to_lds(dst + 4 * i), 0, 0);
    }
#else
    {
      const int kc = tid >> 3;
      const int nb = (tid & 7) * 32;
      const f4* src = (const f4*)(wmat + (size_t)(k_base + kc) * CH + nb);
      f4* dst = (f4*)(B_l + kc * LDB + nb);
#pragma unroll
      for (int i = 0; i < 8; ++i) dst[i] = src[i];
    }
#endif
  };

  // ---- pipelined K loop: C[64x256] += A_im2col[64x768] @ W[768x256] ------
  issue_copy(0, 0);
  for (int ks = 0; ks < NKSTEP; ++ks) {
    const int p = ks & 1;
    if (ks + 1 < NKSTEP) {
      issue_copy(ks + 1, p ^ 1);
#if HAVE_ASYNC
      WAIT_ASYNC(NASY);                     // stage-ks asyncs complete
#endif
#if HAVE_TDM
      WAIT_TENSOR(NTEN);                    // stage-ks TDM complete
#endif
    } else {
#if HAVE_ASYNC
      WAIT_ASYNC(0);
#endif
#if HAVE_TDM
      WAIT_TENSOR(0);
#endif
    }
    __syncthreads();

    const float* A_l = smem + p * BUF;
    const float* B_l = A_l + A_FLOATS;
#pragma unroll
    for (int kk = 0; kk < 32; kk += 4) {    // 8 x (K=4) -> 64 WMMAs
      v2f a[2];
#pragma unroll
      for (int mi = 0; mi < 2; ++mi)
        a[mi] = *(const v2f*)(A_l + (wm * 32 + mi * 16 + l16) * LDA + kk + 2 * hi);
#pragma unroll
      for (int ni = 0; ni < 4; ++ni) {
        const int n = wn * 64 + ni * 16 + l16;
        v2f bf = mk2(B_l[(kk + 2 * hi) * LDB + n], B_l[(kk + 2 * hi + 1) * LDB + n]);
#pragma unroll
        for (int mi = 0; mi < 2; ++mi)
          c[mi][ni] = __builtin_amdgcn_wmma_f32_16x16x4_f32(
              false, a[mi], false, bf, (short)0, c[mi][ni], false, false);
      }
    }
    __syncthreads();
  }

  // ---- Epilogue: spill C to LDS, then fused bias + LayerNorm + ReLU ------
  float* O = smem;                          // 64 x LDO, aliases pipeline bufs
#pragma unroll
  for (int mi = 0; mi < 2; ++mi)
#pragma unroll
    for (int ni = 0; ni < 4; ++ni)
#pragma unroll
      for (int v = 0; v < 8; ++v) {
        const int r   = wm * 32 + mi * 16 + v + 8 * hi;   // C/D VGPR layout
        const int col = wn * 64 + ni * 16 + l16;
        O[r * LDO + col] = c[mi][ni][v];
      }
  __syncthreads();

  float lb0 = 0.f;
  if (PHASE == 2) lb0 = lbp[0];

#pragma unroll
  for (int i = 0; i < 8; ++i) {             // each wave owns 8 rows
    const int r = wv * 8 + i;
    const int m = m_base + r;
    float v[8], s = 0.f, s2 = 0.f;
#pragma unroll
    for (int j = 0; j < 8; ++j) {
      const int col = lane + j * 32;
      const float x = O[r * LDO + col] + cb[col];
      v[j] = x; s += x; s2 += x * x;
    }
#pragma unroll
    for (int off = 16; off > 0; off >>= 1) {   // wave32 butterfly reduce
      s  += __shfl_xor(s,  off, 32);
      s2 += __shfl_xor(s2, off, 32);
    }
    const float mu   = s * (1.f / 256.f);
    const float var  = s2 * (1.f / 256.f) - mu * mu;
    const float rstd = rsqrtf(var + 1e-5f);

    if (PHASE == 1) {
#pragma unroll
      for (int j = 0; j < 8; ++j) {
        const int col = lane + j * 32;
        float y = (v[j] - mu) * rstd * g[col] + beta[col];
        out_h[(size_t)m * CH + col] = fmaxf(y, 0.f);
      }
    } else {
      float d = 0.f;
#pragma unroll
      for (int j = 0; j < 8; ++j) {
        const int col = lane + j * 32;
        float y = (v[j] - mu) * rstd * g[col] + beta[col];
        d += fmaxf(y, 0.f) * lw[col];
      }
#pragma unroll
      for (int off = 16; off > 0; off >>= 1) d += __shfl_xor(d, off, 32);
      if (lane == 0) out_d[m] = fmaxf(d + lb0, 0.f);
    }
  }
}

// ---- inclusive cumsum of durations, one block per batch row ---------------
__global__ __launch_bounds__(512) void cumsum_kernel(const int* __restrict__ dur,
                                                     int* __restrict__ cs) {
  __shared__ int buf[2][SEQL];
  const int b = blockIdx.x, t = threadIdx.x;
  buf[0][t] = dur[b * SEQL + t];
  __syncthreads();
  int src = 0;
  for (int off = 1; off < SEQL; off <<= 1) {
    int v = buf[src][t];
    if (t >= off) v += buf[src][t - off];
    buf[src ^ 1][t] = v;
    src ^= 1;
    __syncthreads();
  }
  cs[b * SEQL + t] = buf[src][t];
}

// ---- length-regulator expand: gather + streaming (nontemporal) stores -----
__global__ __launch_bounds__(256) void expand_kernel(const float* __restrict__ x,
                                                     const int* __restrict__ cs,
                                                     float* __restrict__ out, int M) {
  const int bb   = blockIdx.y;
  const int grp  = threadIdx.x >> 6;          // 4 frames per block
  const int lane = threadIdx.x & 63;          // 64 x float4 = 256 channels
  const int t    = blockIdx.x * 4 + grp;
  if (t >= M) return;
  const int* c = cs + bb * SEQL;
  const int total = c[SEQL - 1];
  int lo = 0, hiB = SEQL;                     // searchsorted(cs, t, 'right')
  while (lo < hiB) {
    const int mid = (lo + hiB) >> 1;
    if (c[mid] <= t) lo = mid + 1; else hiB = mid;
  }
  const int row = lo < (SEQL - 1) ? lo : (SEQL - 1);
  f4 val = (f4){0.f,0.f,0.f,0.f};
  if (t < total)
    val = ((const f4*)(x + ((size_t)bb * SEQL + row) * CH))[lane];
  __builtin_nontemporal_store(val, (f4*)(out + ((size_t)bb * M + t) * CH) + lane);
}

extern "C" void kernel_launch(void* const* d_in, const int* in_sizes, int n_in,
                              void* d_out, int out_size, void* d_ws, size_t ws_size,
                              hipStream_t stream) {
  const float* x    = (const float*)d_in[0];
  const int*   durs = (const int*)d_in[1];
  // d_in[2] (mel_max_length) lives in device memory; derive M from out_size.
  const float* c1w  = (const float*)d_in[3];
  const float* c1b  = (const float*)d_in[4];
  const float* ln1g = (const float*)d_in[5];
  const float* ln1b = (const float*)d_in[6];
  const float* c2w  = (const float*)d_in[7];
  const float* c2b  = (const float*)d_in[8];
  const float* ln2g = (const float*)d_in[9];
  const float* ln2b = (const float*)d_in[10];
  const float* lw   = (const float*)d_in[11];
  const float* lb   = (const float*)d_in[12];

  float* out = (float*)d_out;
  const int M = (out_size - BATCH * SEQL) / (BATCH * CH);   // 4096
  float* durp = out + (size_t)out_size - (size_t)BATCH * SEQL;

  float* h1 = (float*)d_ws;                                  // 16 MiB
  int*   cs = (int*)((char*)d_ws + (size_t)BATCH * SEQL * CH * sizeof(float));

  const int gemm_blocks = (BATCH * SEQL) / 64;               // 256
  conv_ln_kernel<1><<<gemm_blocks, 256, 0, stream>>>(x,  c1w, c1b, ln1g, ln1b,
                                                     nullptr, nullptr, h1, nullptr);
  conv_ln_kernel<2><<<gemm_blocks, 256, 0, stream>>>(h1, c2w, c2b, ln2g, ln2b,
                                                     lw, lb, nullptr, durp);
  cumsum_kernel<<<BATCH, SEQL, 0, stream>>>(durs, cs);
  expand_kernel<<<dim3((M + 3) / 4, BATCH), 256, 0, stream>>>(x, cs, out, M);
}